// HardSPDNorm_60413009985645
// MI455X (gfx1250) — compile-verified
//
#include <hip/hip_runtime.h>

typedef __attribute__((ext_vector_type(2))) float v2f;
typedef __attribute__((ext_vector_type(8))) float v8f;

#define B_      8
#define CF      256
#define HP      128
#define WP      128
#define PIX     (HP * WP)            // 16384
#define NTOT    (B_ * CF * PIX)      // 33554432
#define NIMGDS  (B_ * 3 * PIX)       // 393216
#define NMASKDS (B_ * PIX)           // 131072
#define RED_BLOCKS 1024

// ---------------------------------------------------------------------------
// Kernel 1: 4x4 avg-pool img_p (all 3 channels) + 4x4 max-pool mask (ch 0)
// ---------------------------------------------------------------------------
__global__ __launch_bounds__(256) void pool_kernel(
    const float* __restrict__ img_p,   // (8,3,512,512)
    const float* __restrict__ mask,    // (8,3,512,512)
    float* __restrict__ img_ds,        // (8,3,128,128)
    float* __restrict__ mask_ds)       // (8,128,128)
{
    int tid = blockIdx.x * blockDim.x + threadIdx.x;
    if (tid < NIMGDS) {
        int wo = tid & (WP - 1);
        int ho = (tid >> 7) & (HP - 1);
        int bc = tid >> 14;                       // b*3 + c
        const float* src = img_p + (size_t)bc * 512 * 512;
        float s = 0.0f;
#pragma unroll
        for (int i = 0; i < 4; ++i)
#pragma unroll
            for (int j = 0; j < 4; ++j)
                s += src[(size_t)(ho * 4 + i) * 512 + (wo * 4 + j)];
        img_ds[tid] = s * (1.0f / 16.0f);
    } else {
        int t = tid - NIMGDS;
        if (t < NMASKDS) {
            int wo = t & (WP - 1);
            int ho = (t >> 7) & (HP - 1);
            int b  = t >> 14;
            const float* src = mask + (size_t)(b * 3) * 512 * 512;  // channel 0
            float mx = 0.0f;
#pragma unroll
            for (int i = 0; i < 4; ++i)
#pragma unroll
                for (int j = 0; j < 4; ++j)
                    mx = fmaxf(mx, src[(size_t)(ho * 4 + i) * 512 + (wo * 4 + j)]);
            mask_ds[t] = mx;
        }
    }
}

// ---------------------------------------------------------------------------
// Kernel 2: D_h = 2^-d, d = Chebyshev distance (<=5) to nearest mask==1 pixel
// (collapses the 5-iteration dilate loop into one direct ring scan)
// ---------------------------------------------------------------------------
__global__ __launch_bounds__(256) void dh_kernel(
    const float* __restrict__ mask_ds,  // (8,128,128), values in {0,1}
    float* __restrict__ dh)             // (8,128,128)
{
    int tid = blockIdx.x * blockDim.x + threadIdx.x;
    if (tid >= NMASKDS) return;
    int b = tid >> 14;
    int p = tid & (PIX - 1);
    int y = p >> 7;
    int x = p & (WP - 1);
    const float* mb = mask_ds + (size_t)b * PIX;

    int found = (mb[p] > 0.0f) ? 0 : -1;
    for (int d = 1; d <= 5 && found < 0; ++d) {
        for (int dy = -d; dy <= d && found < 0; ++dy) {
            int yy = y + dy;
            if (yy < 0 || yy > HP - 1) continue;
            const float* row = mb + yy * WP;
            if (dy == -d || dy == d) {
                for (int dx = -d; dx <= d; ++dx) {
                    int xx = x + dx;
                    if (xx < 0 || xx > WP - 1) continue;
                    if (row[xx] > 0.0f) { found = d; break; }
                }
            } else {
                int xl = x - d, xr = x + d;
                if (xl >= 0      && row[xl] > 0.0f) found = d;
                if (xr <= WP - 1 && row[xr] > 0.0f) found = d;
            }
        }
    }
    const float vals[6] = {1.0f, 0.5f, 0.25f, 0.125f, 0.0625f, 0.03125f};
    dh[tid] = (found >= 0) ? vals[found] : 0.0f;
}

// ---------------------------------------------------------------------------
// Kernel 3: partial sums / sums-of-squares over F_in (float4 streaming)
// ---------------------------------------------------------------------------
__global__ __launch_bounds__(256) void reduce_kernel(
    const float* __restrict__ F_in,
    float* __restrict__ partial)        // [RED_BLOCKS] sums, [RED_BLOCKS] sq
{
    __shared__ float ssum[256];
    __shared__ float ssq[256];
    float s = 0.0f, q = 0.0f;
    const float4* F4 = (const float4*)F_in;
    const int n4 = NTOT / 4;
    for (int i = blockIdx.x * blockDim.x + threadIdx.x; i < n4;
         i += gridDim.x * blockDim.x) {
        float4 v = F4[i];
        s += v.x + v.y + v.z + v.w;
        q += v.x * v.x + v.y * v.y + v.z * v.z + v.w * v.w;
    }
    ssum[threadIdx.x] = s;
    ssq[threadIdx.x]  = q;
    __syncthreads();
    for (int off = 128; off > 0; off >>= 1) {
        if (threadIdx.x < off) {
            ssum[threadIdx.x] += ssum[threadIdx.x + off];
            ssq[threadIdx.x]  += ssq[threadIdx.x + off];
        }
        __syncthreads();
    }
    if (threadIdx.x == 0) {
        partial[blockIdx.x]              = ssum[0];
        partial[RED_BLOCKS + blockIdx.x] = ssq[0];
    }
}

// ---------------------------------------------------------------------------
// Kernel 4: finalize mean and scale = 1/sqrt(var^2 + 1e-5)  (note var**2!)
// ---------------------------------------------------------------------------
__global__ __launch_bounds__(256) void finalize_kernel(
    const float* __restrict__ partial,
    float* __restrict__ scalars)        // [0]=mean, [1]=scale
{
    __shared__ float ssum[256];
    __shared__ float ssq[256];
    float s = 0.0f, q = 0.0f;
    for (int i = threadIdx.x; i < RED_BLOCKS; i += 256) {
        s += partial[i];
        q += partial[RED_BLOCKS + i];
    }
    ssum[threadIdx.x] = s;
    ssq[threadIdx.x]  = q;
    __syncthreads();
    for (int off = 128; off > 0; off >>= 1) {
        if (threadIdx.x < off) {
            ssum[threadIdx.x] += ssum[threadIdx.x + off];
            ssq[threadIdx.x]  += ssq[threadIdx.x + off];
        }
        __syncthreads();
    }
    if (threadIdx.x == 0) {
        const float N = (float)NTOT;
        float sum = ssum[0], sq = ssq[0];
        float mean = sum / N;
        float var  = (sq - sum * sum / N) / (N - 1.0f);   // ddof=1
        scalars[0] = mean;
        scalars[1] = rsqrtf(var * var + 1e-5f);
    }
}

// ---------------------------------------------------------------------------
// Kernel 5: fused einsum (V_WMMA_F32_16X16X4_F32) + bias + D_h + normalize.
// One wave handles one (b, 16-channel o-tile) for a 128-pixel group:
//   A (16x4) = gamma_w/beta_w tile, K=3 padded with zero column
//   B (4x16) = img_ds pixel tile
//   D (16x16) = gamma_hp/beta_hp tile, consumed in-register.
// ---------------------------------------------------------------------------
__global__ __launch_bounds__(256) void spdnorm_wmma_kernel(
    const float* __restrict__ F_in,     // (8,256,128,128)
    const float* __restrict__ img_ds,   // (8,3,16384)
    const float* __restrict__ dh,       // (8,16384)
    const float* __restrict__ gamma_w,  // (256,3)
    const float* __restrict__ gamma_b,  // (256)
    const float* __restrict__ beta_w,   // (256,3)
    const float* __restrict__ beta_b,   // (256)
    const float* __restrict__ scalars,  // mean, scale
    float* __restrict__ out)            // (8,256,128,128)
{
    const int wave  = blockIdx.x * (blockDim.x >> 5) + (threadIdx.x >> 5);
    const int lane  = threadIdx.x & 31;
    const int pg    = wave & 127;        // 128-pixel group within batch image
    const int bo    = wave >> 7;         // b*16 + o_tile
    const int b     = bo >> 4;
    const int o_base = (bo & 15) << 4;
    const int m16   = lane & 15;
    const int khalf = lane >> 4;         // lanes 0-15: K=0/1, lanes 16-31: K=2/3

    // A-matrix (16x4) per ISA layout: VGPR0 = K=0|2, VGPR1 = K=1|3 (K=3 -> 0)
    v2f ag, ab;
    {
        int o  = o_base + m16;
        int k0 = khalf * 2;
        ag.x = gamma_w[o * 3 + k0];
        ag.y = khalf ? 0.0f : gamma_w[o * 3 + 1];
        ab.x = beta_w[o * 3 + k0];
        ab.y = khalf ? 0.0f : beta_w[o * 3 + 1];
    }
    const float mean  = scalars[0];
    const float scale = scalars[1];

    const float* imb = img_ds + (size_t)b * 3 * PIX;
    const float* dhb = dh     + (size_t)b * PIX;

    for (int t = 0; t < 8; ++t) {
        const int pix = pg * 128 + t * 16;
        // B-matrix (4x16): rows striped across lanes; VGPR0 = K=0|2, VGPR1 = K=1|3
        v2f bm;
        bm.x = imb[(size_t)(khalf * 2) * PIX + pix + m16];
        bm.y = khalf ? 0.0f : imb[(size_t)PIX + pix + m16];

        v8f zero = {0.0f, 0.0f, 0.0f, 0.0f, 0.0f, 0.0f, 0.0f, 0.0f};
        // D = A x B + 0   (8 args: neg_a, A, neg_b, B, c_mod, C, reuse_a, reuse_b)
        v8f g  = __builtin_amdgcn_wmma_f32_16x16x4_f32(
                     false, ag, false, bm, (short)0, zero, false, false);
        v8f be = __builtin_amdgcn_wmma_f32_16x16x4_f32(
                     false, ab, false, bm, (short)0, zero, false, false);

        const float dhv = dhb[pix + m16];
#pragma unroll
        for (int v = 0; v < 8; ++v) {
            // D layout: lanes 0-15 -> M=v, lanes 16-31 -> M=v+8; N = lane%16
            const int o = o_base + v + khalf * 8;
            const float gm = (g[v]  + gamma_b[o]) * dhv;
            const float bt = (be[v] + beta_b[o])  * dhv;
            const size_t idx = ((size_t)(b * CF + o)) * PIX + pix + m16;
            out[idx] = (F_in[idx] - mean) * scale * gm + bt;
        }
    }
}

// ---------------------------------------------------------------------------
extern "C" void kernel_launch(void* const* d_in, const int* in_sizes, int n_in,
                              void* d_out, int out_size, void* d_ws, size_t ws_size,
                              hipStream_t stream) {
    (void)in_sizes; (void)n_in; (void)out_size; (void)ws_size;
    const float* F_in    = (const float*)d_in[0];
    const float* img_p   = (const float*)d_in[1];
    const float* mask    = (const float*)d_in[2];
    const float* gamma_w = (const float*)d_in[3];
    const float* gamma_b = (const float*)d_in[4];
    const float* beta_w  = (const float*)d_in[5];
    const float* beta_b  = (const float*)d_in[6];
    // d_in[7] = n_ds (=2), d_in[8] = n (=2): behavior hardcoded for these.
    float* out = (float*)d_out;

    float* ws      = (float*)d_ws;
    float* img_ds  = ws;                     // 393216 floats
    float* mask_ds = img_ds + NIMGDS;        // 131072 floats
    float* dh      = mask_ds + NMASKDS;      // 131072 floats
    float* partial = dh + NMASKDS;           // 2*RED_BLOCKS floats
    float* scalars = partial + 2 * RED_BLOCKS; // 2 floats

    // 1) downsample img + mask
    pool_kernel<<<(NIMGDS + NMASKDS + 255) / 256, 256, 0, stream>>>(
        img_p, mask, img_ds, mask_ds);
    // 2) distance-decay map (replaces 5 dilation passes)
    dh_kernel<<<NMASKDS / 256, 256, 0, stream>>>(mask_ds, dh);
    // 3) global mean/var pass over F_in (leaves F_in hot in the 192MB L2)
    reduce_kernel<<<RED_BLOCKS, 256, 0, stream>>>(F_in, partial);
    finalize_kernel<<<1, 256, 0, stream>>>(partial, scalars);
    // 4) fused WMMA einsum + normalize + apply (16384 waves)
    spdnorm_wmma_kernel<<<2048, 256, 0, stream>>>(
        F_in, img_ds, dh, gamma_w, gamma_b, beta_w, beta_b, scalars, out);
}